// SynthLocalLoss_65670049956015
// MI455X (gfx1250) — compile-verified
//
#include <hip/hip_runtime.h>
#include <hip/hip_bf16.h>
#include <math.h>

typedef __attribute__((ext_vector_type(2))) float v2f;
typedef __attribute__((ext_vector_type(8))) float v8f;

#define GRID_B 4
#define GRID_Z 32
#define GRID_Y 512
#define GRID_X 512

#define SLOTS_LOG2 20
#define SLOTS (1u << SLOTS_LOG2)
#define SLOT_MASK (SLOTS - 1u)
#define EMPTY_KEY (-1)
#define BLOCK 256
#define NSTREAM 16   // 4 live streams (occ/off/feat/cnt) + 12 zero pad -> uniform WMMA loads

__device__ __forceinline__ unsigned slot_of(int h) {
    // Knuth multiplicative hash, take top SLOTS_LOG2 bits
    return ((unsigned)h * 2654435761u) >> (32 - SLOTS_LOG2);
}

// ---------------- kernel 1: init hash table (key,val packed as int2) ----------------
__global__ void init_table_kernel(int2* __restrict__ table, int slots) {
    int i = blockIdx.x * blockDim.x + threadIdx.x;
    if (i < slots) {
        table[i] = make_int2(EMPTY_KEY, 0x7FFFFFFF);   // global_store_b64
    }
}

// ---------------- kernel 1b: zero the padded partials buffer ----------------
__global__ void zero_partials_kernel(float* __restrict__ partials, int n) {
    int i = blockIdx.x * blockDim.x + threadIdx.x;
    if (i < n) partials[i] = 0.f;
}

// ---------------- kernel 2: insert lidar voxels ----------------
// Stable-argsort + searchsorted-left semantics == MIN original lidar row index
// per voxel hash (atomicMin => order-independent => deterministic).
__global__ void insert_lidar_kernel(const int* __restrict__ lidar_idx, int Nl,
                                    int2* __restrict__ table) {
    int i = blockIdx.x * blockDim.x + threadIdx.x;
    if (i >= Nl) return;
    const int4 li = ((const int4*)lidar_idx)[i];       // global_load_b128
    int h = ((li.x * GRID_Z + li.y) * GRID_Y + li.z) * GRID_X + li.w;  // < 2^25, never -1
    unsigned s = slot_of(h);
    for (;;) {
        int* cell = (int*)&table[s];
        int old = atomicCAS(cell, EMPTY_KEY, h);
        if (old == EMPTY_KEY || old == h) {
            atomicMin(cell + 1, i);
            break;
        }
        s = (s + 1) & SLOT_MASK;
    }
}

__device__ __forceinline__ int table_lookup(const int2* __restrict__ table, int h) {
    unsigned s = slot_of(h);
    for (;;) {
        int2 kv = table[s];                 // one global_load_b64 per probe
        if (kv.x == h) return kv.y;
        if (kv.x == EMPTY_KEY) return -1;
        s = (s + 1) & SLOT_MASK;
    }
}

// ---------------- kernel 3: per-radar matching + loss terms ----------------
__global__ void radar_loss_kernel(const float* __restrict__ pred_feat,
                                  const float* __restrict__ pred_occ,
                                  const float* __restrict__ lidar_feat,
                                  const int* __restrict__ radar_idx,
                                  const int2* __restrict__ table,
                                  int Nr, int NB4,
                                  float* __restrict__ partials) {
    int r = blockIdx.x * blockDim.x + threadIdx.x;
    float occ_term = 0.f, off_term = 0.f, feat_term = 0.f, cnt_term = 0.f;

    if (r < Nr) {
        const int4   ri = ((const int4*)radar_idx)[r];   // global_load_b128
        const float4 pf = ((const float4*)pred_feat)[r]; // global_load_b128
        const float  occ = pred_occ[r];
        int b = ri.x, z = ri.y, y = ri.z, x = ri.w;

        // Prefetch all 27 candidate slot lines so the random L2 probes overlap.
        #pragma unroll
        for (int dz = -1; dz <= 1; ++dz)
            #pragma unroll
            for (int dy = -1; dy <= 1; ++dy)
                #pragma unroll
                for (int dx = -1; dx <= 1; ++dx) {
                    int zz = min(max(z + dz, 0), GRID_Z - 1);
                    int yy = min(max(y + dy, 0), GRID_Y - 1);
                    int xx = min(max(x + dx, 0), GRID_X - 1);
                    int h  = ((b * GRID_Z + zz) * GRID_Y + yy) * GRID_X + xx;
                    __builtin_prefetch(&table[slot_of(h)], 0, 3);  // global_prefetch
                }

        int best_dl1 = 1000000000;
        int best_row = -1;
        float bdz = 0.f, bdy = 0.f, bdx = 0.f;

        // Loop order matches meshgrid('ij') flatten: dz slowest, dx fastest.
        // Strict '<' keeps the FIRST k achieving the minimum (jnp.argmin).
        #pragma unroll
        for (int dz = -1; dz <= 1; ++dz)
            #pragma unroll
            for (int dy = -1; dy <= 1; ++dy)
                #pragma unroll
                for (int dx = -1; dx <= 1; ++dx) {
                    int zz = min(max(z + dz, 0), GRID_Z - 1);
                    int yy = min(max(y + dy, 0), GRID_Y - 1);
                    int xx = min(max(x + dx, 0), GRID_X - 1);
                    int h  = ((b * GRID_Z + zz) * GRID_Y + yy) * GRID_X + xx;
                    int row = table_lookup(table, h);
                    if (row >= 0) {
                        // dL1 uses the UNCLIPPED offsets
                        int dl1 = (dz < 0 ? -dz : dz) + (dy < 0 ? -dy : dy) + (dx < 0 ? -dx : dx);
                        if (dl1 < best_dl1) {
                            best_dl1 = dl1;
                            best_row = row;
                            bdz = (float)(zz - z);   // gt_offsets use CLIPPED diffs
                            bdy = (float)(yy - y);
                            bdx = (float)(xx - x);
                        }
                    }
                }

        float t = (best_row >= 0) ? 1.f : 0.f;
        // softplus(x) = max(x,0) + log1p(exp(-|x|))  (stable logaddexp(x,0))
        occ_term = (fmaxf(occ, 0.f) + log1pf(expf(-fabsf(occ)))) - occ * t;

        if (best_row >= 0) {
            float d0 = pf.x - bdz, d1 = pf.y - bdy, d2 = pf.z - bdx;
            float a0 = fabsf(d0), a1 = fabsf(d1), a2 = fabsf(d2);
            off_term = (a0 < 1.f ? 0.5f * d0 * d0 : a0 - 0.5f)
                     + (a1 < 1.f ? 0.5f * d1 * d1 : a1 - 0.5f)
                     + (a2 < 1.f ? 0.5f * d2 * d2 : a2 - 0.5f);
            feat_term = fabsf(pf.w - lidar_feat[best_row * 16]);  // only column 0 used
            cnt_term = 1.f;
        }
    }

    // deterministic fixed-order block tree reduction
    __shared__ float s0[BLOCK], s1[BLOCK], s2[BLOCK], s3[BLOCK];
    int tid = threadIdx.x;
    s0[tid] = occ_term; s1[tid] = off_term; s2[tid] = feat_term; s3[tid] = cnt_term;
    __syncthreads();
    for (int stride = BLOCK / 2; stride > 0; stride >>= 1) {
        if (tid < stride) {
            s0[tid] += s0[tid + stride];
            s1[tid] += s1[tid + stride];
            s2[tid] += s2[tid + stride];
            s3[tid] += s3[tid + stride];
        }
        __syncthreads();
    }
    if (tid == 0) {
        partials[0 * NB4 + blockIdx.x] = s0[0];
        partials[1 * NB4 + blockIdx.x] = s1[0];
        partials[2 * NB4 + blockIdx.x] = s2[0];
        partials[3 * NB4 + blockIdx.x] = s3[0];
    }
}

// ---------------- kernel 4: WMMA final reduction (one wave32) ----------------
// A = ones(16x4): D = A*B + C column-sums B over K. Streams occ/off/feat/cnt in
// columns 0..3; columns 4..15 read pre-zeroed pad streams so EVERY lane does one
// unconditional v2f (b64) load per step -> straight-line load/wmma loop.
// B layout (32-bit 4x16): element [k][n] in lane (k>=2 ? 16+n : n), VGPR (k&1).
__global__ void final_reduce_kernel(const float* __restrict__ partials, int NB4, int Nr,
                                    float* __restrict__ out) {
    int lane = threadIdx.x;            // 0..31, single wave
    int n = lane & 15;                 // column / stream index
    int khi = (lane >= 16) ? 2 : 0;
    const float* p = partials + n * NB4 + khi;  // 8B aligned (NB4 % 4 == 0)

    v2f a;  a[0] = 1.f; a[1] = 1.f;    // A[m][k] = 1 for all m,k
    v8f c = {};

    for (int base = 0; base < NB4; base += 4) {
        v2f bmat = *(const v2f*)(p + base);     // global_load_b64, no guards
        c = __builtin_amdgcn_wmma_f32_16x16x4_f32(
                /*neg_a=*/false, a, /*neg_b=*/false, bmat,
                /*c_mod=*/(short)0, c, /*reuse_a=*/false, /*reuse_b=*/false);
    }

    // D VGPR0, lanes 0..15 hold row M=0: D[0][n] = total of stream n
    __shared__ float sums[16];
    if (lane < 16) sums[lane] = c[0];
    __syncthreads();
    if (lane == 0) {
        float occ_sum  = sums[0];
        float off_sum  = sums[1];
        float feat_sum = sums[2];
        float cntf     = sums[3];
        float occ_loss  = occ_sum / (float)Nr;
        float off_loss  = off_sum / fmaxf(cntf * 3.0f, 1.0f);
        float feat_loss = feat_sum / fmaxf(cntf, 1.0f);
        out[0] = 0.2f * occ_loss + 1.0f * off_loss + 1.0f * feat_loss;
    }
}

extern "C" void kernel_launch(void* const* d_in, const int* in_sizes, int n_in,
                              void* d_out, int out_size, void* d_ws, size_t ws_size,
                              hipStream_t stream) {
    const float* pred_feat  = (const float*)d_in[0];   // (Nr,4)
    const float* pred_occ   = (const float*)d_in[1];   // (Nr,1)
    const float* lidar_feat = (const float*)d_in[2];   // (Nl,16)
    const int*   radar_idx  = (const int*)d_in[3];     // (Nr,4)
    const int*   lidar_idx  = (const int*)d_in[4];     // (Nl,4)

    int Nr = in_sizes[1];          // pred_occ has Nr elements
    int Nl = in_sizes[4] / 4;      // lidar_indices (Nl,4)
    float* out = (float*)d_out;

    int NB  = (Nr + BLOCK - 1) / BLOCK;
    int NB4 = (NB + 3) & ~3;       // pad to multiple of 4 WMMA K-steps

    // workspace layout: [table: SLOTS x int2 (8MB)][partials: NSTREAM*NB4 f32]
    int2*  table    = (int2*)d_ws;
    float* partials = (float*)(table + SLOTS);
    int    npart    = NSTREAM * NB4;

    init_table_kernel<<<(int)((SLOTS + BLOCK - 1) / BLOCK), BLOCK, 0, stream>>>(table, (int)SLOTS);
    zero_partials_kernel<<<(npart + BLOCK - 1) / BLOCK, BLOCK, 0, stream>>>(partials, npart);
    insert_lidar_kernel<<<(Nl + BLOCK - 1) / BLOCK, BLOCK, 0, stream>>>(lidar_idx, Nl, table);
    radar_loss_kernel<<<NB, BLOCK, 0, stream>>>(pred_feat, pred_occ, lidar_feat, radar_idx,
                                                table, Nr, NB4, partials);
    final_reduce_kernel<<<1, 32, 0, stream>>>(partials, NB4, Nr, out);
}